// InfBallBounded_62637803045535
// MI455X (gfx1250) — compile-verified
//
#include <hip/hip_runtime.h>
#include <math.h>

#define EPS      0.031f
#define N_FEAT   3072
#define M_FEAT   3072
#define NBATCH   16
#define WAVES_PB 4
#define BLOCK_T  (WAVES_PB * 32)
#define UNROLL   8

typedef float v2f __attribute__((ext_vector_type(2)));
typedef float v8f __attribute__((ext_vector_type(8)));

// Pass 1: per-(batch, m-tile, n-slice) WMMA reduction using
//   zu = c.nu + r.|nu|,  zl = c.nu - r.|nu|
//   acc_c row0 = sum_n c[n]*nu[n][m]      (A row0 = c, B = nu)
//   acc_r row1 = sum_n r[n]*|nu[n][m]|    (A row1 = r, B = |nu|)
// DIRECT==1: write zl/zu straight to d_out. DIRECT==0: write (cdot,rdot) partials.
template <int NSLICE, int DIRECT>
__global__ __launch_bounds__(BLOCK_T)
void ibp_wmma_pass1(const float* __restrict__ X,
                    const float* __restrict__ nu,
                    float* __restrict__ out)
{
    // [0..3] = zero pad (A rows 2..15), then c[NSLICE], then r[NSLICE]
    __shared__ float smem[4 + 2 * NSLICE];
    float* const zpad = smem;
    float* const cbuf = smem + 4;
    float* const rbuf = cbuf + NSLICE;

    const int b     = blockIdx.y;
    const int s     = blockIdx.z;
    const int nbase = s * NSLICE;
    const int tid   = threadIdx.x;

    if (tid < 4) zpad[tid] = 0.0f;

    // Stage c = (l+u)/2, r = (u-l)/2 into LDS.
    const float* xb = X + (size_t)b * N_FEAT + nbase;
    for (int i = tid; i < NSLICE; i += BLOCK_T) {
        float x = xb[i];
        float l = fmaxf(x - EPS, 0.0f);
        float u = fminf(x + EPS, 1.0f);
        cbuf[i] = 0.5f * (l + u);
        rbuf[i] = 0.5f * (u - l);
    }
    __syncthreads();

    const int wave = tid >> 5;
    const int lane = tid & 31;
    const int half = lane >> 4;    // ISA fp32 layout: K = 2*half + vgpr (A and B)
    const int sub  = lane & 15;    // A row (M) / B col (N)

    const int mbase = (blockIdx.x * WAVES_PB + wave) * 16;
    const int col   = mbase + sub;

    // B element pointer: rows nbase + 4k + 2*half (+1 in .y), column `col`.
    const float* nup =
        nu + ((size_t)b * N_FEAT + (size_t)nbase + 2 * half) * M_FEAT + col;

    // Branch-free A source: lanes sub==0 -> c pairs, sub==1 -> r pairs,
    // lanes sub>=2 -> zero pad with zero stride (always a plain ds_load_b64).
    const float* aptr;
    int aoff, asel;
    if (sub == 0)      { aptr = cbuf; aoff = 2 * half; asel = 1; }
    else if (sub == 1) { aptr = rbuf; aoff = 2 * half; asel = 1; }
    else               { aptr = zpad; aoff = 0;        asel = 0; }

    v8f acc_c = {0.f, 0.f, 0.f, 0.f, 0.f, 0.f, 0.f, 0.f};
    v8f acc_r = {0.f, 0.f, 0.f, 0.f, 0.f, 0.f, 0.f, 0.f};

    auto loadG = [&](int g, v2f* bb, v2f* aa) {
        #pragma unroll
        for (int u = 0; u < UNROLL; ++u) {
            const int c4 = (g * UNROLL + u) * 4;
            const float* p = nup + (size_t)c4 * (size_t)M_FEAT;
            v2f t;
            t.x = __builtin_nontemporal_load(p);            // row c4 + 2*half
            t.y = __builtin_nontemporal_load(p + M_FEAT);   // row c4 + 2*half + 1
            bb[u] = t;
            aa[u] = *(const v2f*)(aptr + aoff + asel * c4); // ds_load_b64
        }
    };
    auto consume = [&](const v2f* bb, const v2f* aa) {
        #pragma unroll
        for (int u = 0; u < UNROLL; ++u) {
            v2f ba = { fabsf(bb[u].x), fabsf(bb[u].y) };
            acc_c = __builtin_amdgcn_wmma_f32_16x16x4_f32(
                        false, aa[u], false, bb[u], (short)0, acc_c, false, false);
            acc_r = __builtin_amdgcn_wmma_f32_16x16x4_f32(
                        false, aa[u], false, ba,    (short)0, acc_r, false, false);
        }
    };

    constexpr int CHUNKS = NSLICE / 4;          // K=4 per WMMA pair
    constexpr int NG     = CHUNKS / UNROLL;     // 12 (split) or 96 (direct)
    static_assert(NG % 2 == 0, "ping-pong needs even group count");

    // Distance-1 software pipeline, ping-pong buffers (no rotation copies).
    v2f b0[UNROLL], a0[UNROLL], b1[UNROLL], a1[UNROLL];
    loadG(0, b0, a0);
    #pragma unroll 1
    for (int g = 0; g < NG; g += 2) {
        loadG(g + 1, b1, a1);
        consume(b0, a0);
        if (g + 2 < NG) loadG(g + 2, b0, a0);
        consume(b1, a1);
    }

    // C/D layout: VGPR0 lanes0-15 = row M=0 (c-dot), VGPR1 lanes0-15 = row M=1 (r-dot).
    const float cd = acc_c[0];
    const float rd = acc_r[1];

    if (lane < 16) {
        const int gm = mbase + lane;
        if (DIRECT) {
            out[(size_t)b * M_FEAT + gm]            = cd - rd;   // zl
            out[(size_t)(NBATCH + b) * M_FEAT + gm] = cd + rd;   // zu
        } else {
            float* wsp = out + (((size_t)s * NBATCH + b) * 2) * M_FEAT;
            wsp[gm]          = cd;
            wsp[M_FEAT + gm] = rd;
        }
    }
}

// Pass 2: deterministic fixed-order combine of the n-split partials.
__global__ __launch_bounds__(256)
void ibp_reduce(const float* __restrict__ ws, float* __restrict__ out, int nsplit)
{
    const int idx = blockIdx.x * blockDim.x + threadIdx.x;
    if (idx >= NBATCH * M_FEAT) return;
    const int b = idx / M_FEAT;
    const int m = idx - b * M_FEAT;

    float cd = 0.f, rd = 0.f;
    for (int s = 0; s < nsplit; ++s) {
        const float* p = ws + (((size_t)s * NBATCH + b) * 2) * M_FEAT;
        cd += p[m];
        rd += p[M_FEAT + m];
    }
    out[(size_t)b * M_FEAT + m]            = cd - rd;   // zl
    out[(size_t)(NBATCH + b) * M_FEAT + m] = cd + rd;   // zu
}

extern "C" void kernel_launch(void* const* d_in, const int* in_sizes, int n_in,
                              void* d_out, int out_size, void* d_ws, size_t ws_size,
                              hipStream_t stream)
{
    const float* X  = (const float*)d_in[0];   // (16,3,32,32) fp32
    const float* nu = (const float*)d_in[1];   // (16,3072,3072) fp32
    float* out = (float*)d_out;                // [zl | zu], 2*16*3072 fp32

    constexpr int NSPLIT = 8;
    const size_t need = (size_t)NSPLIT * NBATCH * 2 * M_FEAT * sizeof(float); // 3 MB

    dim3 block(BLOCK_T, 1, 1);
    if (ws_size >= need) {
        float* ws = (float*)d_ws;
        dim3 grid(M_FEAT / (16 * WAVES_PB), NBATCH, NSPLIT);
        ibp_wmma_pass1<N_FEAT / NSPLIT, 0><<<grid, block, 0, stream>>>(X, nu, ws);
        const int tot = NBATCH * M_FEAT;
        ibp_reduce<<<(tot + 255) / 256, 256, 0, stream>>>(ws, out, NSPLIT);
    } else {
        dim3 grid(M_FEAT / (16 * WAVES_PB), NBATCH, 1);
        ibp_wmma_pass1<N_FEAT, 1><<<grid, block, 0, stream>>>(X, nu, out);
    }
}